// APLayer_52656299049563
// MI455X (gfx1250) — compile-verified
//
#include <hip/hip_runtime.h>
#include <hip/hip_bf16.h>
#include <math.h>

typedef __attribute__((ext_vector_type(2))) float v2f;
typedef __attribute__((ext_vector_type(8))) float v8f;

#define DFEAT 64

// ---------------------------------------------------------------------------
// Kernel 1: ew_table[t] = edge_weight[t,:] . edge_weight_param + 1
// ---------------------------------------------------------------------------
__global__ void k_ewtable(const float* __restrict__ ew,
                          const float* __restrict__ ewp,
                          float* __restrict__ table, int ntypes) {
  int t = threadIdx.x;
  if (t < ntypes) {
    float s = 1.0f;
    for (int j = 0; j < ntypes; ++j) s = fmaf(ew[t * ntypes + j], ewp[j], s);
    table[t] = s;
  }
}

// ---------------------------------------------------------------------------
// Kernel 2: per-node attention logit + zero-mask via V_WMMA_F32_16X16X4_F32.
// One wave32 handles 16 nodes. A = 16x4 feat tile (documented f32 A layout:
// lane<16 -> M=lane, K = k0 + {0,1}; lane>=16 -> M=lane-16, K = k0 + {2,3}).
// B = attn chunk replicated across all 16 columns (K mapping mirrors A / the
// 16-bit B rule: V0 = K0(lo)/K2(hi), V1 = K1(lo)/K3(hi)).
// cabs uses B = ones (layout-invariant) for the row abs-sums.
// C is column-constant: lane 0 holds dots for nodes m0..m0+7 in c[0..7],
// lane 16 holds m0+8..m0+15. EXEC kept all-ones: loads clamped, stores guarded.
// ---------------------------------------------------------------------------
__global__ void __launch_bounds__(256) k_node_logits(
    const float* __restrict__ feat, const float* __restrict__ attn,
    float* __restrict__ nw_exp, int n_nodes) {
  const int lane  = threadIdx.x & 31;
  const int wave  = threadIdx.x >> 5;
  const int h     = lane >> 4;   // lane half
  const int m     = lane & 15;   // row within tile
  const int node0 = (blockIdx.x * 8 + wave) * 16;

  int node = node0 + m;
  int nc   = node < n_nodes ? node : (n_nodes - 1);
  const float* __restrict__ frow = feat + (size_t)nc * DFEAT;

  v8f cdot = {};
  v8f cabs = {};
#pragma unroll
  for (int k0 = 0; k0 < DFEAT; k0 += 4) {
    const int kk = k0 + 2 * h;
    v2f a, aa, b, ones;
    a.x  = frow[kk];
    a.y  = frow[kk + 1];
    aa.x = __builtin_fabsf(a.x);
    aa.y = __builtin_fabsf(a.y);
    b.x  = attn[kk];
    b.y  = attn[kk + 1];
    ones.x = 1.0f;
    ones.y = 1.0f;
    // (neg_a, A, neg_b, B, c_mod, C, reuse_a, reuse_b)
    cdot = __builtin_amdgcn_wmma_f32_16x16x4_f32(false, a, false, b,
                                                 (short)0, cdot, false, false);
    cabs = __builtin_amdgcn_wmma_f32_16x16x4_f32(false, aa, false, ones,
                                                 (short)0, cabs, false, false);
  }

  if (m == 0) {  // lanes 0 and 16 write 8 nodes each
    const int base = node0 + 8 * h;
#pragma unroll
    for (int r = 0; r < 8; ++r) {
      int nd = base + r;
      if (nd < n_nodes) {
        // nw_exp = exp(nw / T) * node_mask, T == 1
        nw_exp[nd] = (cabs[r] == 0.0f) ? 0.0f : expf(cdot[r]);
      }
    }
  }
}

// ---------------------------------------------------------------------------
// Kernel 3: edge scatter. One wave32 per edge; lane l gathers feat[src][2l..2l+1]
// (coalesced 256B float2 read, served out of L2 since feat fits in 192MB L2)
// and issues two non-returning global_atomic_add_f32 into ft[dst]. Lane 0 also
// accumulates the softmax normalizer.
// ---------------------------------------------------------------------------
__global__ void __launch_bounds__(256) k_edge_scatter(
    const float* __restrict__ feat, const int* __restrict__ src,
    const int* __restrict__ dst, const int* __restrict__ efeat,
    const float* __restrict__ nw_exp, const float* __restrict__ table,
    float* __restrict__ nw_sum, float* __restrict__ ft, int n_edges) {
  const long tid = (long)blockIdx.x * blockDim.x + threadIdx.x;
  const int e    = (int)(tid >> 5);   // uniform within a wave
  const int lane = (int)(tid & 31);
  if (e >= n_edges) return;

  const int s = src[e];
  const int d = dst[e];
  const float w = nw_exp[s];                 // broadcast load (L2-resident)
  if (lane == 0) atomicAdd(&nw_sum[d], w);

  const float scale = w * table[efeat[e] - 1];

  const float2* __restrict__ fs = (const float2*)(feat + (size_t)s * DFEAT);
  float2 v = fs[lane];
  float* __restrict__ fd = ft + (size_t)d * DFEAT + 2 * lane;
  atomicAdd(fd,     v.x * scale);
  atomicAdd(fd + 1, v.y * scale);
}

// ---------------------------------------------------------------------------
// Kernel 4: finalize. One wave32 per node; recompute the zero-mask with a
// shuffle reduction (order-independent: sum of |x| is 0 iff all zero),
// normalize, and blend with the original features.
// ---------------------------------------------------------------------------
__global__ void __launch_bounds__(256) k_finalize(
    const float* __restrict__ feat, const float* __restrict__ ft,
    const float* __restrict__ nw_sum, float* __restrict__ out, int n_nodes) {
  const long tid  = (long)blockIdx.x * blockDim.x + threadIdx.x;
  const int node  = (int)(tid >> 5);
  const int lane  = (int)(tid & 31);
  if (node >= n_nodes) return;

  const float2* __restrict__ f2 = (const float2*)(feat + (size_t)node * DFEAT);
  float2 f = f2[lane];

  float a = __builtin_fabsf(f.x) + __builtin_fabsf(f.y);
#pragma unroll
  for (int off = 16; off > 0; off >>= 1) a += __shfl_xor(a, off, 32);
  const bool masked = (a != 0.0f);  // node_mask == 1 -> keep original feat

  const float ns    = nw_sum[node];
  const float denom = (ns < 1e-8f) ? 1.0f : ns;

  const float2* __restrict__ r2 = (const float2*)(ft + (size_t)node * DFEAT);
  float2 r = r2[lane];

  float2 o;
  o.x = masked ? f.x : (r.x / denom);
  o.y = masked ? f.y : (r.y / denom);
  ((float2*)out)[(size_t)node * (DFEAT / 2) + lane] = o;
}

// ---------------------------------------------------------------------------
extern "C" void kernel_launch(void* const* d_in, const int* in_sizes, int n_in,
                              void* d_out, int out_size, void* d_ws,
                              size_t ws_size, hipStream_t stream) {
  const float* feat  = (const float*)d_in[0];
  const float* attn  = (const float*)d_in[1];
  const float* ew    = (const float*)d_in[2];
  const float* ewp   = (const float*)d_in[3];
  const int*   src   = (const int*)d_in[4];
  const int*   dst   = (const int*)d_in[5];
  const int*   efeat = (const int*)d_in[6];
  float*       out   = (float*)d_out;

  const int N = in_sizes[0] / DFEAT;   // 100000
  const int E = in_sizes[4];           // 1000000
  const int T = in_sizes[3];           // 8 edge types

  // Workspace layout (f32): [table:64][nw_exp:N][nw_sum:N][ft:N*64] ~= 26.4 MB
  float* ws     = (float*)d_ws;
  float* table  = ws;
  float* nw_exp = ws + 64;
  float* nw_sum = nw_exp + N;
  float* ft     = nw_sum + N;

  // Accumulators must be zeroed every call (graph-capture-safe memsets).
  hipMemsetAsync(nw_sum, 0, (size_t)N * sizeof(float), stream);
  hipMemsetAsync(ft, 0, (size_t)N * DFEAT * sizeof(float), stream);

  k_ewtable<<<1, 32, 0, stream>>>(ew, ewp, table, T);

  const int nodeBlocks = (N + 127) / 128;  // 8 waves/block * 16 nodes/wave
  k_node_logits<<<nodeBlocks, 256, 0, stream>>>(feat, attn, nw_exp, N);

  const long eThreads = (long)E * 32;
  const int  eBlocks  = (int)((eThreads + 255) / 256);
  k_edge_scatter<<<eBlocks, 256, 0, stream>>>(feat, src, dst, efeat, nw_exp,
                                              table, nw_sum, ft, E);

  const long fThreads = (long)N * 32;
  const int  fBlocks  = (int)((fThreads + 255) / 256);
  k_finalize<<<fBlocks, 256, 0, stream>>>(feat, ft, nw_sum, out, N);
}